// SFC_CAE_47313359733334
// MI455X (gfx1250) — compile-verified
//
#include <hip/hip_runtime.h>
#include <hip/hip_bf16.h>
#include <math.h>

typedef __attribute__((ext_vector_type(2))) float v2f;
typedef __attribute__((ext_vector_type(8))) float v8f;

#define BATCH 16
#define LAT   16
#define HID   512
#define NCOL  131104          // SFC * NFC * START = 2*16*4097
#define NTILE 8194            // NCOL / 16
#define NN    262144
#define LDSTR 514             // padded LDS row stride for h1 (breaks bank conflicts)

// ---------------- fc1: [16,16] @ [16,512] + bias, tanh ----------------
__global__ void fc1_k(const float* __restrict__ x, const float* __restrict__ w,
                      const float* __restrict__ bias, float* __restrict__ h1) {
    int idx = blockIdx.x * blockDim.x + threadIdx.x;     // 16*512
    if (idx >= BATCH * HID) return;
    int j = idx & (HID - 1);
    int b = idx >> 9;
    float s = bias[j];
#pragma unroll
    for (int k = 0; k < LAT; ++k) s += x[b * LAT + k] * w[k * HID + j];
    h1[idx] = tanhf(s);
}

// -------- fc2: [16,512] @ [512,131104] + bias, tanh  (WMMA f32 16x16x4) --------
// One wave computes one 16(M=batch) x 16(N) output tile; K looped in steps of 4.
__global__ __launch_bounds__(256) void fc2_wmma_k(const float* __restrict__ h1,
                                                  const float* __restrict__ w,
                                                  const float* __restrict__ bias,
                                                  float* __restrict__ h2) {
    __shared__ float sA[BATCH * LDSTR];
    int tid = threadIdx.x;
    // stage h1 (16x512) into LDS with padded stride
    for (int i = tid; i < BATCH * HID; i += 256) {
        int m = i >> 9, k = i & (HID - 1);
        sA[m * LDSTR + k] = h1[i];
    }
    __syncthreads();

    int wave = tid >> 5;
    int lane = tid & 31;
    int tile = blockIdx.x * 8 + wave;
    if (tile >= NTILE) return;               // whole-wave uniform exit: EXEC stays full

    int n0    = tile * 16;
    int nlane = lane & 15;                   // N within tile / M for A-frag
    int half  = lane >> 4;

    v8f acc = {};
    const float* wp = w + n0 + nlane;        // B[k][n] = w[k*NCOL + n]
    const float* ap = sA + nlane * LDSTR;    // A[m][k] (m = batch row)
#pragma unroll 4
    for (int k = 0; k < HID; k += 4) {
        int ka = k + 2 * half;
        v2f a, b;
        a.x = ap[ka];
        a.y = ap[ka + 1];
        b.x = wp[(long)ka * NCOL];
        b.y = wp[(long)(ka + 1) * NCOL];
        acc = __builtin_amdgcn_wmma_f32_16x16x4_f32(
            /*neg_a=*/false, a, /*neg_b=*/false, b,
            /*c_mod=*/(short)0, acc, /*reuse_a=*/false, /*reuse_b=*/false);
    }
    float bf = bias[n0 + nlane];
    float* op = h2 + n0 + nlane;             // h2[b][n], b = v + 8*half
#pragma unroll
    for (int v = 0; v < 8; ++v) {
        int m = v + 8 * half;
        op[(long)m * NCOL] = tanhf(acc[v] + bf);
    }
}

// ---------------- generic ConvTranspose1d (K=32,S=4,P=16) + tanh ----------------
// y[b,co,t] = bias[co] + sum_ci sum_{j=0..7, i=itop-j in [0,Lin)} x[b,ci,i]*w[ci,co,r+4j]
// where u=t+16, r=u&3, itop=u>>2.  Input indexed with (stride, offset) so layer 0
// can deinterleave h2's [b, s*2 + isfc] layout directly.
__global__ __launch_bounds__(256) void convt_k(const float* __restrict__ in,
                                               const float* __restrict__ w,
                                               const float* __restrict__ bias,
                                               float* __restrict__ out,
                                               int Cin, int Cout, int Lin, int Lout,
                                               int estride, int eoff, long bstride) {
    long idx   = (long)blockIdx.x * 256 + threadIdx.x;
    long total = (long)BATCH * Cout * Lout;
    if (idx >= total) return;
    int t  = (int)(idx % Lout);
    int co = (int)((idx / Lout) % Cout);
    int b  = (int)(idx / ((long)Lout * Cout));

    int u    = t + 16;
    int r    = u & 3;
    int itop = u >> 2;
    float s = bias[co];
    for (int ci = 0; ci < Cin; ++ci) {
        const float* ip = in + b * bstride + (long)ci * Lin * estride + eoff;
        const float* wp = w + (ci * Cout + co) * 32 + r;
#pragma unroll
        for (int j = 0; j < 8; ++j) {
            int i = itop - j;
            if (i >= 0 && i < Lin) s += ip[(long)i * estride] * wp[4 * j];
        }
    }
    out[idx] = tanhf(s);
}

// ------- fused: per-branch 3-neighbor SFC smoothing + final 2-branch combine -------
__global__ __launch_bounds__(256) void spfinal_k(const float* __restrict__ c0,
                                                 const float* __restrict__ c1,
                                                 const float* __restrict__ w0, const float* __restrict__ b0,
                                                 const int* __restrict__ m0, const int* __restrict__ o0, const int* __restrict__ p0,
                                                 const float* __restrict__ w1, const float* __restrict__ b1,
                                                 const int* __restrict__ m1, const int* __restrict__ o1, const int* __restrict__ p1,
                                                 const float* __restrict__ fw, const float* __restrict__ fb,
                                                 float* __restrict__ out) {
    int n = blockIdx.x * 256 + threadIdx.x;
    int b = blockIdx.y;
    const float* r0 = c0 + (long)b * NN;
    const float* r1 = c1 + (long)b * NN;
    float z0 = tanhf(r0[m0[n]] * w0[n * 3 + 0] +
                     r0[o0[n]] * w0[n * 3 + 1] +
                     r0[p0[n]] * w0[n * 3 + 2] + b0[n]);
    float z1 = tanhf(r1[m1[n]] * w1[n * 3 + 0] +
                     r1[o1[n]] * w1[n * 3 + 1] +
                     r1[p1[n]] * w1[n * 3 + 2] + b1[n]);
    out[(long)b * NN + n] = tanhf(z0 * fw[n * 2] + z1 * fw[n * 2 + 1] + fb[n]);
}

extern "C" void kernel_launch(void* const* d_in, const int* in_sizes, int n_in,
                              void* d_out, int out_size, void* d_ws, size_t ws_size,
                              hipStream_t stream) {
    const float* x      = (const float*)d_in[0];
    const float* fc1_w  = (const float*)d_in[1];
    const float* fc1_b  = (const float*)d_in[2];
    const float* fc2_w  = (const float*)d_in[3];
    const float* fc2_b  = (const float*)d_in[4];
    // per-branch blocks of 11 inputs starting at 5 and 16
    const float* ct_w[2][3]; const float* ct_b[2][3];
    const float* sp_w[2]; const float* sp_b[2];
    const int *ordi[2], *mini[2], *plui[2];
    for (int i = 0; i < 2; ++i) {
        int base = 5 + i * 11;
        for (int j = 0; j < 3; ++j) {
            ct_w[i][j] = (const float*)d_in[base + 2 * j];
            ct_b[i][j] = (const float*)d_in[base + 2 * j + 1];
        }
        sp_w[i] = (const float*)d_in[base + 6];
        sp_b[i] = (const float*)d_in[base + 7];
        ordi[i] = (const int*)d_in[base + 8];
        mini[i] = (const int*)d_in[base + 9];
        plui[i] = (const int*)d_in[base + 10];
    }
    const float* fsp_w = (const float*)d_in[27];
    const float* fsp_b = (const float*)d_in[28];
    float* out = (float*)d_out;

    // workspace layout (floats)
    float* ws = (float*)d_ws;
    float* H1 = ws;                       // 16*512          =     8,192
    float* H2 = H1 + 8192;                // 16*131104       = 2,097,664
    float* A  = H2 + 2097664;             // 16*8*16385      = 2,097,280
    float* Bb = A  + 2097280;             // 16*4*65537      = 4,194,368
    float* C0 = Bb + 4194368;             // 16*262144       = 4,194,304
    float* C1 = C0 + 4194304;             // 16*262144       = 4,194,304
    (void)ws_size; (void)n_in; (void)in_sizes; (void)out_size;

    // fc1
    fc1_k<<<(BATCH * HID + 255) / 256, 256, 0, stream>>>(x, fc1_w, fc1_b, H1);
    // fc2 (WMMA): 8194 tiles, 8 waves/block
    fc2_wmma_k<<<(NTILE + 7) / 8, 256, 0, stream>>>(H1, fc2_w, fc2_b, H2);

    float* Cout[2] = {C0, C1};
    for (int i = 0; i < 2; ++i) {
        // layer 0: Cin=16,Cout=8, 4097 -> 16385, deinterleaved read of H2
        long t0 = (long)BATCH * 8 * 16385;
        convt_k<<<(unsigned)((t0 + 255) / 256), 256, 0, stream>>>(
            H2, ct_w[i][0], ct_b[i][0], A, 16, 8, 4097, 16385, 2, i, (long)NCOL);
        // layer 1: Cin=8,Cout=4, 16385 -> 65537
        long t1 = (long)BATCH * 4 * 65537;
        convt_k<<<(unsigned)((t1 + 255) / 256), 256, 0, stream>>>(
            A, ct_w[i][1], ct_b[i][1], Bb, 8, 4, 16385, 65537, 1, 0, (long)8 * 16385);
        // layer 2: Cin=4,Cout=1, 65537 -> 262144
        long t2 = (long)BATCH * NN;
        convt_k<<<(unsigned)((t2 + 255) / 256), 256, 0, stream>>>(
            Bb, ct_w[i][2], ct_b[i][2], Cout[i], 4, 1, 65537, NN, 1, 0, (long)4 * 65537);
    }

    dim3 g(NN / 256, BATCH);
    spfinal_k<<<g, 256, 0, stream>>>(C0, C1,
                                     sp_w[0], sp_b[0], mini[0], ordi[0], plui[0],
                                     sp_w[1], sp_b[1], mini[1], ordi[1], plui[1],
                                     fsp_w, fsp_b, out);
}